// LightGCN_23854248362838
// MI455X (gfx1250) — compile-verified
//
#include <hip/hip_runtime.h>

#define EMB_DIM 128
#define WPB 8              // waves per block (wave32)
#define TPB (WPB * 32)
#define AHEAD 3            // async gathers in flight (4 LDS slots)

// ---- CDNA5 async global->LDS path (gfx1250), with safe fallback -------------
#if defined(__has_builtin)
#if __has_builtin(__builtin_amdgcn_global_load_async_to_lds_b128)
#define HAVE_ASYNC_LDS 1
#endif
#endif
#ifndef HAVE_ASYNC_LDS
#define HAVE_ASYNC_LDS 0
#endif

typedef int v4i __attribute__((ext_vector_type(4)));
typedef __attribute__((address_space(1))) v4i g_v4i;   // global int4
typedef __attribute__((address_space(3))) v4i l_v4i;   // LDS int4

// ---- SpMM: y[rows[e]] += vals[e] * x[cols[e]] (atomic scatter) --------------
// One wave per contiguous edge chunk; lane l owns float4 at column 4*l
// (512B fully-coalesced row per edge). Indices via scalar loads (wave-uniform).
__global__ __launch_bounds__(TPB) void spmm_atomic(
    const float* __restrict__ x, float* __restrict__ y,
    const int* __restrict__ rows, const int* __restrict__ cols,
    const float* __restrict__ vals, int nedges, int chunk)
{
    const int lane = threadIdx.x & 31;
    const int w    = threadIdx.x >> 5;
    // wave-uniform wave id -> scalar (s_load) index fetches
    const int gw = __builtin_amdgcn_readfirstlane((int)(blockIdx.x * WPB + w));

    int beg = gw * chunk;
    if (beg >= nedges) return;
    int end = beg + chunk;
    if (end > nedges) end = nedges;

#if HAVE_ASYNC_LDS
    // 4-slot ring of async gathers: slot = e & 3, 512B each
    __shared__ float4 buf[WPB][4][32];

    // prologue: fill the pipeline
    {
        int pre = beg + AHEAD;
        if (pre > end) pre = end;
        for (int e = beg; e < pre; ++e) {
            int c = cols[e];
            __builtin_amdgcn_global_load_async_to_lds_b128(
                (g_v4i*)(x + (long)c * EMB_DIM + lane * 4),
                (l_v4i*)&buf[w][e & 3][lane], 0, 0);
        }
    }

    int e     = beg;
    int emain = end - AHEAD;   // may be < beg (short chunk) -> main loop skipped
    for (; e < emain; ++e) {
        int cn = cols[e + AHEAD];
        __builtin_amdgcn_global_load_async_to_lds_b128(
            (g_v4i*)(x + (long)cn * EMB_DIM + lane * 4),
            (l_v4i*)&buf[w][(e + AHEAD) & 3][lane], 0, 0);
        __builtin_amdgcn_s_wait_asynccnt(AHEAD);   // oldest (edge e) has landed

        const float  v  = vals[e];
        const long   r  = rows[e];
        const float4 xv = buf[w][e & 3][lane];     // ds_load_b128
        float* dst = y + r * EMB_DIM + lane * 4;
        atomicAdd(dst + 0, v * xv.x);              // global_atomic_add_f32
        atomicAdd(dst + 1, v * xv.y);
        atomicAdd(dst + 2, v * xv.z);
        atomicAdd(dst + 3, v * xv.w);
    }
    // tail: drain remaining (<= AHEAD) edges
    for (; e < end; ++e) {
        __builtin_amdgcn_s_wait_asynccnt(0);
        const float  v  = vals[e];
        const long   r  = rows[e];
        const float4 xv = buf[w][e & 3][lane];
        float* dst = y + r * EMB_DIM + lane * 4;
        atomicAdd(dst + 0, v * xv.x);
        atomicAdd(dst + 1, v * xv.y);
        atomicAdd(dst + 2, v * xv.z);
        atomicAdd(dst + 3, v * xv.w);
    }
#else
    for (int e = beg; e < end; ++e) {
        const long   c  = cols[e];
        const float  v  = vals[e];
        const long   r  = rows[e];
        const float4 xv = *(const float4*)(x + c * EMB_DIM + lane * 4);
        float* dst = y + r * EMB_DIM + lane * 4;
        atomicAdd(dst + 0, v * xv.x);
        atomicAdd(dst + 1, v * xv.y);
        atomicAdd(dst + 2, v * xv.z);
        atomicAdd(dst + 3, v * xv.w);
    }
#endif
}

// ---- elementwise helpers (float4 vectorized, grid-stride) -------------------
__global__ void k_init(const float4* __restrict__ emb, float4* __restrict__ acc,
                       float4* __restrict__ A, float4* __restrict__ B, int n4)
{
    const float4 z = {0.f, 0.f, 0.f, 0.f};
    int i = blockIdx.x * blockDim.x + threadIdx.x;
    int s = gridDim.x * blockDim.x;
    for (; i < n4; i += s) { acc[i] = emb[i]; A[i] = z; B[i] = z; }
}

__global__ void k_accum(float4* __restrict__ acc, const float4* __restrict__ src,
                        float4* __restrict__ zbuf, int n4)
{
    const float4 z = {0.f, 0.f, 0.f, 0.f};
    int i = blockIdx.x * blockDim.x + threadIdx.x;
    int s = gridDim.x * blockDim.x;
    for (; i < n4; i += s) {
        float4 a = acc[i]; const float4 b = src[i];
        a.x += b.x; a.y += b.y; a.z += b.z; a.w += b.w;
        acc[i] = a;
        if (zbuf) zbuf[i] = z;   // re-zero next scatter target (fused)
    }
}

__global__ void k_accum_scale(float4* __restrict__ acc, const float4* __restrict__ src,
                              float sc, int n4)
{
    int i = blockIdx.x * blockDim.x + threadIdx.x;
    int s = gridDim.x * blockDim.x;
    for (; i < n4; i += s) {
        float4 a = acc[i]; const float4 b = src[i];
        a.x = (a.x + b.x) * sc; a.y = (a.y + b.y) * sc;
        a.z = (a.z + b.z) * sc; a.w = (a.w + b.w) * sc;
        acc[i] = a;
    }
}

// ---- launcher ---------------------------------------------------------------
extern "C" void kernel_launch(void* const* d_in, const int* in_sizes, int n_in,
                              void* d_out, int out_size, void* d_ws, size_t ws_size,
                              hipStream_t stream)
{
    const float* emb  = (const float*)d_in[0];
    const int*   rows = (const int*)d_in[1];
    const int*   cols = (const int*)d_in[2];
    const float* vals = (const float*)d_in[3];

    const long N      = (long)in_sizes[0] / EMB_DIM;   // 100000 nodes
    const int  nedges = in_sizes[1];                   // 2M edges
    const int  n4     = (int)(N * (EMB_DIM / 4));

    float* acc = (float*)d_out;                        // running accumulator
    float* A   = (float*)d_ws;                         // layer ping
    float* B   = A + (size_t)N * EMB_DIM;              // layer pong (102.4MB total)

    const int eb = 2048;                               // elementwise blocks
    const int sb = 4096;                               // spmm blocks -> 32768 waves
    const int nwaves = sb * WPB;
    const int chunk  = (nedges + nwaves - 1) / nwaves; // ~61 edges/wave, host-side

    // acc = emb ; A = 0 ; B = 0
    k_init<<<eb, 256, 0, stream>>>((const float4*)emb, (float4*)acc,
                                   (float4*)A, (float4*)B, n4);
    // layer 1: A = S*emb ; acc += A
    spmm_atomic<<<sb, TPB, 0, stream>>>(emb, A, rows, cols, vals, nedges, chunk);
    k_accum<<<eb, 256, 0, stream>>>((float4*)acc, (const float4*)A, nullptr, n4);
    // layer 2: B = S*A ; acc += B ; A = 0
    spmm_atomic<<<sb, TPB, 0, stream>>>(A, B, rows, cols, vals, nedges, chunk);
    k_accum<<<eb, 256, 0, stream>>>((float4*)acc, (const float4*)B, (float4*)A, n4);
    // layer 3: A = S*B ; acc = (acc + A) * 0.25
    spmm_atomic<<<sb, TPB, 0, stream>>>(B, A, rows, cols, vals, nedges, chunk);
    k_accum_scale<<<eb, 256, 0, stream>>>((float4*)acc, (const float4*)A, 0.25f, n4);
}